// GINModel_56487409877356
// MI455X (gfx1250) — compile-verified
//
#include <hip/hip_runtime.h>

#define N_NODES 50000
#define N_EDGES 640000
#define N_GRAPHS 64
#define HDIM 128

typedef __attribute__((ext_vector_type(2))) float v2f;
typedef __attribute__((ext_vector_type(8))) float v8f;

// D(16x16,f32) = A(16x4,f32) x B(4x16,f32) + C   — full fp32, matches reference numerics
__device__ __forceinline__ v8f wmma4(v2f a, v2f b, v8f c) {
  return __builtin_amdgcn_wmma_f32_16x16x4_f32(false, a, false, b, (short)0, c, false, false);
}

__global__ void zero_f4_kernel(float4* __restrict__ p, int n4) {
  int i = blockIdx.x * blockDim.x + threadIdx.x;
  if (i < n4) { float4 z = {0.f, 0.f, 0.f, 0.f}; p[i] = z; }
}

// scatter-add: agg[dst[e]] += h[src[e]]  (F floats per node, 4 per thread)
template<int F>
__global__ void scatter_add_kernel(const float* __restrict__ h,
                                   const int* __restrict__ src,
                                   const int* __restrict__ dst,
                                   float* __restrict__ agg) {
  const int QUADS = F / 4;
  int tid = blockIdx.x * blockDim.x + threadIdx.x;
  if (tid >= N_EDGES * QUADS) return;
  int e = tid / QUADS;
  int q = tid - e * QUADS;
  int s = src[e], d = dst[e];
  const float4 val = *(const float4*)(h + (size_t)s * F + q * 4);
  float* base = agg + (size_t)d * F + q * 4;
  unsafeAtomicAdd(base + 0, val.x);
  unsafeAtomicAdd(base + 1, val.y);
  unsafeAtomicAdd(base + 2, val.z);
  unsafeAtomicAdd(base + 3, val.w);
}

// Fused GIN MLP: hout = relu( relu((x+agg)@W1 + b1) @ W2 + b2 )
// block = 256 threads (8 waves), 32 node rows per block.
// wave w owns output columns [16w,16w+16), two 16-row tiles, fp32 WMMA.
template<int KIN>
__global__ __launch_bounds__(256) void gin_mlp_kernel(
    const float* __restrict__ xin,   // [N,KIN]
    const float* __restrict__ agg,   // [N,KIN]
    const float* __restrict__ W1,    // [KIN,128]
    const float* __restrict__ B1,    // [128]
    const float* __restrict__ W2,    // [128,128]
    const float* __restrict__ B2,    // [128]
    float* __restrict__ hout)        // [N,128]
{
  const int ROWS = 32;
  const int PAD = 4;  // LDS row padding -> conflict-free strided A-fragment reads
  __shared__ float sA[ROWS][KIN + PAD];
  __shared__ float sMid[ROWS][HDIM + PAD];
  __shared__ float sB1[HDIM];
  __shared__ float sB2[HDIM];

  const int t = threadIdx.x;
  const int wid = t >> 5;      // wave id 0..7 (wave32)
  const int lane = t & 31;
  const int row0 = blockIdx.x * ROWS;

  // stage sA = x + agg (GIN eps=0: h_in = x + sum-aggregation)
  for (int i = t; i < ROWS * KIN; i += 256) {
    int r = i / KIN, c = i - r * KIN;
    int gr = row0 + r;
    float val = 0.f;
    if (gr < N_NODES) val = xin[(size_t)gr * KIN + c] + agg[(size_t)gr * KIN + c];
    sA[r][c] = val;
  }
  if (t < HDIM) { sB1[t] = B1[t]; sB2[t] = B2[t]; }
  __syncthreads();

  const int n0 = wid * 16;     // this wave's output column base
  const int mn = lane & 15;    // A: row m ; B/D: col n
  const int kh = lane >> 4;    // lane-half selects K (A/B) / row+8 (D)

  // ---- matmul1: sMid = relu(sA @ W1 + b1), K = KIN ----
  {
    v8f acc0 = {}; v8f acc1 = {};
#pragma unroll
    for (int kk = 0; kk < KIN / 4; ++kk) {
      int k = kk * 4 + 2 * kh;
      v2f b;
      b.x = W1[(size_t)k * HDIM + n0 + mn];
      b.y = W1[(size_t)(k + 1) * HDIM + n0 + mn];
      v2f a0, a1;
      a0.x = sA[mn][k];      a0.y = sA[mn][k + 1];
      a1.x = sA[16 + mn][k]; a1.y = sA[16 + mn][k + 1];
      acc0 = wmma4(a0, b, acc0);
      acc1 = wmma4(a1, b, acc1);
    }
    float bias = sB1[n0 + mn];
#pragma unroll
    for (int e = 0; e < 8; ++e) {
      int rr = e + 8 * kh;
      sMid[rr][n0 + mn]      = fmaxf(acc0[e] + bias, 0.f);
      sMid[16 + rr][n0 + mn] = fmaxf(acc1[e] + bias, 0.f);
    }
  }
  __syncthreads();

  // ---- matmul2: hout = relu(sMid @ W2 + b2), K = 128 ----
  {
    v8f acc0 = {}; v8f acc1 = {};
#pragma unroll
    for (int kk = 0; kk < HDIM / 4; ++kk) {
      int k = kk * 4 + 2 * kh;
      v2f b;
      b.x = W2[(size_t)k * HDIM + n0 + mn];
      b.y = W2[(size_t)(k + 1) * HDIM + n0 + mn];
      v2f a0, a1;
      a0.x = sMid[mn][k];      a0.y = sMid[mn][k + 1];
      a1.x = sMid[16 + mn][k]; a1.y = sMid[16 + mn][k + 1];
      acc0 = wmma4(a0, b, acc0);
      acc1 = wmma4(a1, b, acc1);
    }
    float bias = sB2[n0 + mn];
#pragma unroll
    for (int e = 0; e < 8; ++e) {
      int rr = e + 8 * kh;
      int gr0 = row0 + rr;
      int gr1 = row0 + 16 + rr;
      if (gr0 < N_NODES) hout[(size_t)gr0 * HDIM + n0 + mn] = fmaxf(acc0[e] + bias, 0.f);
      if (gr1 < N_NODES) hout[(size_t)gr1 * HDIM + n0 + mn] = fmaxf(acc1[e] + bias, 0.f);
    }
  }
}

// global_add_pool of hcat=[h1|h2|h3] per graph: pooled[batch[n]][0:384] += ...
__global__ void pool_kernel(const float* __restrict__ h1, const float* __restrict__ h2,
                            const float* __restrict__ h3, const int* __restrict__ batch,
                            float* __restrict__ pooled) {
  const int Q = 96;  // 384 / 4
  int tid = blockIdx.x * blockDim.x + threadIdx.x;
  if (tid >= N_NODES * Q) return;
  int nidx = tid / Q;
  int q = tid - nidx * Q;
  int f = q * 4;  // 0..380
  const float* h = (f < 128) ? h1 : ((f < 256) ? h2 : h3);
  int fl = f & 127;
  const float4 val = *(const float4*)(h + (size_t)nidx * HDIM + fl);
  float* base = pooled + (size_t)batch[nidx] * 384 + f;
  unsafeAtomicAdd(base + 0, val.x);
  unsafeAtomicAdd(base + 1, val.y);
  unsafeAtomicAdd(base + 2, val.z);
  unsafeAtomicAdd(base + 3, val.w);
}

// classifier: relu(BN(pooled @ cW1 + cb1)) @ cW2 + cb2   (one block per graph)
__global__ __launch_bounds__(256) void classifier_kernel(
    const float* __restrict__ pooled,  // [64,384]
    const float* __restrict__ W1,      // [384,256]
    const float* __restrict__ B1,      // [256]
    const float* __restrict__ gamma, const float* __restrict__ beta,
    const float* __restrict__ mean, const float* __restrict__ var,
    const float* __restrict__ W2,      // [256,4]
    const float* __restrict__ B2,      // [4]
    float* __restrict__ out)           // [64,4]
{
  __shared__ float sp[384];
  __shared__ float sz[256];
  int g = blockIdx.x;
  int t = threadIdx.x;
  for (int i = t; i < 384; i += 256) sp[i] = pooled[(size_t)g * 384 + i];
  __syncthreads();
  float acc = B1[t];
  for (int k = 0; k < 384; ++k) acc += sp[k] * W1[(size_t)k * 256 + t];
  acc = (acc - mean[t]) * rsqrtf(var[t] + 1e-5f) * gamma[t] + beta[t];
  sz[t] = fmaxf(acc, 0.f);
  __syncthreads();
  if (t < 4) {
    float o = B2[t];
    for (int j = 0; j < 256; ++j) o += sz[j] * W2[(size_t)j * 4 + t];
    out[(size_t)g * 4 + t] = o;
  }
}

static inline int cdiv(long a, long b) { return (int)((a + b - 1) / b); }

extern "C" void kernel_launch(void* const* d_in, const int* in_sizes, int n_in,
                              void* d_out, int out_size, void* d_ws, size_t ws_size,
                              hipStream_t stream) {
  (void)in_sizes; (void)n_in; (void)out_size; (void)ws_size;

  const float* x     = (const float*)d_in[0];
  const int*   ei    = (const int*)d_in[1];
  const int*   batch = (const int*)d_in[2];
  const float* l1W1 = (const float*)d_in[3];
  const float* l1b1 = (const float*)d_in[4];
  const float* l1W2 = (const float*)d_in[5];
  const float* l1b2 = (const float*)d_in[6];
  const float* l2W1 = (const float*)d_in[7];
  const float* l2b1 = (const float*)d_in[8];
  const float* l2W2 = (const float*)d_in[9];
  const float* l2b2 = (const float*)d_in[10];
  const float* l3W1 = (const float*)d_in[11];
  const float* l3b1 = (const float*)d_in[12];
  const float* l3W2 = (const float*)d_in[13];
  const float* l3b2 = (const float*)d_in[14];
  const float* cW1  = (const float*)d_in[15];
  const float* cb1  = (const float*)d_in[16];
  const float* bng  = (const float*)d_in[17];
  const float* bnb  = (const float*)d_in[18];
  const float* bnm  = (const float*)d_in[19];
  const float* bnv  = (const float*)d_in[20];
  const float* cW2  = (const float*)d_in[21];
  const float* cb2  = (const float*)d_in[22];

  const int* src = ei;             // edge_index[0]
  const int* dst = ei + N_EDGES;   // edge_index[1]

  // workspace layout (floats): agg | h1 | h2 | h3 | pooled   (~102.5 MB)
  float* ws = (float*)d_ws;
  size_t nh = (size_t)N_NODES * HDIM;
  float* agg    = ws;
  float* h1     = ws + nh;
  float* h2     = h1 + nh;
  float* h3     = h2 + nh;
  float* pooled = h3 + nh;

  dim3 blk(256);
  const int mlp_grid = cdiv(N_NODES, 32);

  // ---- layer 1 (K_in = 8) ----
  zero_f4_kernel<<<cdiv((long)N_NODES * 8 / 4, 256), blk, 0, stream>>>((float4*)agg, N_NODES * 8 / 4);
  scatter_add_kernel<8><<<cdiv((long)N_EDGES * 2, 256), blk, 0, stream>>>(x, src, dst, agg);
  gin_mlp_kernel<8><<<mlp_grid, blk, 0, stream>>>(x, agg, l1W1, l1b1, l1W2, l1b2, h1);

  // ---- layer 2 (K_in = 128) ----
  zero_f4_kernel<<<cdiv((long)nh / 4, 256), blk, 0, stream>>>((float4*)agg, (int)(nh / 4));
  scatter_add_kernel<128><<<cdiv((long)N_EDGES * 32, 256), blk, 0, stream>>>(h1, src, dst, agg);
  gin_mlp_kernel<128><<<mlp_grid, blk, 0, stream>>>(h1, agg, l2W1, l2b1, l2W2, l2b2, h2);

  // ---- layer 3 (K_in = 128) ----
  zero_f4_kernel<<<cdiv((long)nh / 4, 256), blk, 0, stream>>>((float4*)agg, (int)(nh / 4));
  scatter_add_kernel<128><<<cdiv((long)N_EDGES * 32, 256), blk, 0, stream>>>(h2, src, dst, agg);
  gin_mlp_kernel<128><<<mlp_grid, blk, 0, stream>>>(h2, agg, l3W1, l3b1, l3W2, l3b2, h3);

  // ---- jk='cat' + global_add_pool ----
  zero_f4_kernel<<<cdiv((long)N_GRAPHS * 384 / 4, 256), blk, 0, stream>>>((float4*)pooled, N_GRAPHS * 384 / 4);
  pool_kernel<<<cdiv((long)N_NODES * 96, 256), blk, 0, stream>>>(h1, h2, h3, batch, pooled);

  // ---- classifier ----
  classifier_kernel<<<N_GRAPHS, blk, 0, stream>>>(pooled, cW1, cb1, bng, bnb, bnm, bnv, cW2, cb2,
                                                  (float*)d_out);
}